// GraphLayers_8624294330605
// MI455X (gfx1250) — compile-verified
//
#include <hip/hip_runtime.h>
#include <hip/hip_bf16.h>

typedef float v2f __attribute__((ext_vector_type(2)));
typedef float v8f __attribute__((ext_vector_type(8)));

// ---------------------------------------------------------------- zero buffer
__global__ void zero_f32(float* __restrict__ p, long long n) {
    long long i = (long long)blockIdx.x * blockDim.x + threadIdx.x;
    long long stride = (long long)gridDim.x * blockDim.x;
    for (; i < n; i += stride) p[i] = 0.0f;
}

// ---------------------------------------------------------------- degrees
__global__ void degree_kernel(const int* __restrict__ src, const int* __restrict__ dst,
                              float* __restrict__ deg_out, float* __restrict__ deg_in, int E) {
    int gid = blockIdx.x * blockDim.x + threadIdx.x;
    if (gid < E) {
        atomicAdd(&deg_out[src[gid]], 1.0f);
        atomicAdd(&deg_in[dst[gid]], 1.0f);
    }
}

// ---------------------------------------------------------------- deg -> deg^-0.5 (in place)
__global__ void norm_kernel(float* __restrict__ deg, int n) {
    int gid = blockIdx.x * blockDim.x + threadIdx.x;
    if (gid < n) {
        float d = deg[gid];
        d = d < 1.0f ? 1.0f : d;
        deg[gid] = rsqrtf(d);
    }
}

// ---------------------------------------------------------------- h = (x * norm_src) @ W  via V_WMMA_F32_16X16X4_F32
// One wave = one 16-row tile, 4 accumulators covering the 64 output columns.
// A 16x4 f32 frag: lane l -> row m=l&15, K = k0 + 2*(l>>4) + {0,1}   (float2 global load)
// B 4x16  f32 frag: lane l -> col n=l&15, K = k0 + 2*(l>>4) + {0,1}
//   W staged into LDS swizzled by K-pair: sW[pair=k>>1][n*2 + (k&1)],
//   so every B fragment is ONE contiguous 8B ds_load_b64.
// C/D 16x16 f32: VGPR v -> row 8*(l>>4)+v, col l&15
__global__ __launch_bounds__(256) void gemm_wmma(const float* __restrict__ x,
                                                 const float* __restrict__ W,
                                                 const float* __restrict__ norm_src,
                                                 float* __restrict__ h,
                                                 int nNodes, int nTiles) {
    __shared__ float sW[64 * 128];  // 32 KB: [pair 0..63][n*2 + parity]

    const int tid = threadIdx.x;
    {   // stage + swizzle W: 1024 units of (pair, 4-col chunk); 256 threads x 4
#pragma unroll
        for (int i = 0; i < 4; ++i) {
            const int u    = tid + 256 * i;
            const int pair = u >> 4;          // 0..63  -> k = 2*pair, 2*pair+1
            const int n4   = (u & 15) * 4;    // 0,4,...,60
            const int k    = pair * 2;
            const float4 a = *(const float4*)(W + (size_t)k * 64 + n4);
            const float4 b = *(const float4*)(W + (size_t)(k + 1) * 64 + n4);
            float4* d = (float4*)(sW + (size_t)pair * 128 + n4 * 2);
            d[0] = make_float4(a.x, b.x, a.y, b.y);
            d[1] = make_float4(a.z, b.z, a.w, b.w);
        }
    }
    __syncthreads();

    const int wave = tid >> 5;
    const int lane = tid & 31;
    const int tile = blockIdx.x * 8 + wave;
    if (tile >= nTiles) return;                    // wave-uniform: EXEC stays all-1s

    const int row0 = tile * 16;
    const int m    = lane & 15;
    const int half = lane >> 4;

    int row = row0 + m;
    if (row >= nNodes) row = nNodes - 1;           // clamp (keeps EXEC full)
    const float  ns   = norm_src[row];
    const float* xrow = x + (size_t)row * 128;

    v8f acc0 = {}, acc1 = {}, acc2 = {}, acc3 = {};

#pragma unroll 4
    for (int k0 = 0; k0 < 128; k0 += 4) {
        const int kk = k0 + 2 * half;
        // A fragment (scaled by row norm)
        const float2 av = *(const float2*)(xrow + kk);
        v2f a; a.x = av.x * ns; a.y = av.y * ns;
        // B fragments: contiguous float2 per fragment from swizzled LDS
        const v2f* wrow = (const v2f*)(sW + (size_t)((k0 >> 1) + half) * 128) + m;
        const v2f b0 = wrow[0];      // n = m
        const v2f b1 = wrow[16];     // n = 16 + m
        const v2f b2 = wrow[32];     // n = 32 + m
        const v2f b3 = wrow[48];     // n = 48 + m
        acc0 = __builtin_amdgcn_wmma_f32_16x16x4_f32(false, a, false, b0, (short)0, acc0, false, false);
        acc1 = __builtin_amdgcn_wmma_f32_16x16x4_f32(false, a, false, b1, (short)0, acc1, false, false);
        acc2 = __builtin_amdgcn_wmma_f32_16x16x4_f32(false, a, false, b2, (short)0, acc2, false, false);
        acc3 = __builtin_amdgcn_wmma_f32_16x16x4_f32(false, a, false, b3, (short)0, acc3, false, false);
    }

    // store D: VGPR v -> row 8*half+v, col n = m (+16*j per accumulator)
    if (row0 + 16 <= nNodes) {
        // wave-uniform fast path: no per-lane predication, stores can clause
        float* hr = h + (size_t)(row0 + 8 * half) * 64 + m;
#pragma unroll
        for (int v = 0; v < 8; ++v, hr += 64) {
            hr[0]  = acc0[v];
            hr[16] = acc1[v];
            hr[32] = acc2[v];
            hr[48] = acc3[v];
        }
    } else {
#pragma unroll
        for (int v = 0; v < 8; ++v) {
            const int r = row0 + 8 * half + v;
            if (r < nNodes) {
                float* hr = h + (size_t)r * 64 + m;
                hr[0]  = acc0[v];
                hr[16] = acc1[v];
                hr[32] = acc2[v];
                hr[48] = acc3[v];
            }
        }
    }
}

// ---------------------------------------------------------------- edge scatter-add: out[dst] += h[src]
// 16 lanes per edge, float4 per lane -> 256B coalesced gather per edge.
__global__ void scatter_kernel(const float* __restrict__ h, const int* __restrict__ src,
                               const int* __restrict__ dst, float* __restrict__ out,
                               long long total) {
    long long gid = (long long)blockIdx.x * blockDim.x + threadIdx.x;
    if (gid >= total) return;
    const int e  = (int)(gid >> 4);
    const int c4 = (int)(gid & 15);
    const int s  = src[e];
    const int d  = dst[e];
    const float4 v = *(const float4*)(h + (size_t)s * 64 + c4 * 4);
    float* o = out + (size_t)d * 64 + c4 * 4;
    atomicAdd(o + 0, v.x);
    atomicAdd(o + 1, v.y);
    atomicAdd(o + 2, v.z);
    atomicAdd(o + 3, v.w);
}

// ---------------------------------------------------------------- out = out * norm_dst + bias
__global__ void finish_kernel(float* __restrict__ out, const float* __restrict__ norm_dst,
                              const float* __restrict__ bias, long long total) {
    long long gid = (long long)blockIdx.x * blockDim.x + threadIdx.x;
    if (gid >= total) return;
    const int node = (int)(gid >> 6);
    const int c    = (int)(gid & 63);
    out[gid] = out[gid] * norm_dst[node] + bias[c];
}

extern "C" void kernel_launch(void* const* d_in, const int* in_sizes, int n_in,
                              void* d_out, int out_size, void* d_ws, size_t ws_size,
                              hipStream_t stream) {
    const float* x    = (const float*)d_in[0];  // [N,128]
    const float* W    = (const float*)d_in[1];  // [128,64]
    const float* bias = (const float*)d_in[2];  // [64]
    const int*   src  = (const int*)d_in[3];    // [E]
    const int*   dst  = (const int*)d_in[4];    // [E]

    const int IN_C = 128, OUT_C = 64;
    const int nNodes = in_sizes[0] / IN_C;
    const int E      = in_sizes[3];

    float* out     = (float*)d_out;
    float* deg_out = (float*)d_ws;            // [nNodes] -> becomes norm_src
    float* deg_in  = deg_out + nNodes;        // [nNodes] -> becomes norm_dst
    float* h       = deg_in + nNodes;         // [nNodes * 64]

    // zero accumulation targets (harness does not re-zero between replays)
    zero_f32<<<2048, 256, 0, stream>>>(out, (long long)out_size);
    zero_f32<<<512, 256, 0, stream>>>(deg_out, (long long)2 * nNodes);

    degree_kernel<<<(E + 255) / 256, 256, 0, stream>>>(src, dst, deg_out, deg_in, E);
    norm_kernel<<<(2 * nNodes + 255) / 256, 256, 0, stream>>>(deg_out, 2 * nNodes);

    const int nTiles = (nNodes + 15) / 16;    // 6250 for N=100000 (exact)
    gemm_wmma<<<(nTiles + 7) / 8, 256, 0, stream>>>(x, W, deg_out, h, nNodes, nTiles);

    const long long totE = (long long)E * 16;
    scatter_kernel<<<(int)((totE + 255) / 256), 256, 0, stream>>>(h, src, dst, out, totE);

    const long long totO = (long long)nNodes * OUT_C;
    finish_kernel<<<(int)((totO + 255) / 256), 256, 0, stream>>>(out, deg_in, bias, totO);
}